// GCN_15126874816523
// MI455X (gfx1250) — compile-verified
//
#include <hip/hip_runtime.h>

// ---------------------------------------------------------------------------
// GCN 2-layer forward for MI455X (gfx1250, wave32).
//   layer: out = D^-1/2 (A+I) D^-1/2 (X W) + b
// GEMMs use V_WMMA_F32_16X16X4_F32 (full fp32, matches fp32 reference).
// Aggregation is HBM-bound scatter-add with global_atomic_add_f32.
// ---------------------------------------------------------------------------

typedef __attribute__((ext_vector_type(2))) float v2f;
typedef __attribute__((ext_vector_type(8))) float v8f;

#define IN_CH  256
#define HID    128
#define OUT_CH 64

// -------------------------- degree / normalization -------------------------

__global__ void k_init_deg(float* __restrict__ deg, int n) {
    int i = blockIdx.x * blockDim.x + threadIdx.x;
    if (i < n) deg[i] = 1.0f;  // self-loop contributes 1
}

__global__ void k_count_deg(const long long* __restrict__ dst,
                            float* __restrict__ deg, int E) {
    int i = blockIdx.x * blockDim.x + threadIdx.x;
    if (i < E) atomicAdd(&deg[(int)dst[i]], 1.0f);
}

__global__ void k_deg_to_dinv(float* __restrict__ deg, int n) {
    int i = blockIdx.x * blockDim.x + threadIdx.x;
    if (i < n) deg[i] = rsqrtf(deg[i]);  // deg >= 1 always (self-loops)
}

// ------------------------------ WMMA GEMM ----------------------------------
// H[n x NOUT] = X[n x K] * W[K x NOUT]
// One wave computes a 16-row stripe across all NOUT columns.
// A 16x4 f32 frag : lane l -> row (l&15), K = kk + (l>>4)*2 + {0,1}
// B  4x16 f32 frag: lane l -> col (l&15), K = kk + (l>>4)*2 + {0,1}
// D 16x16 f32     : vgpr v, lane l -> row v + 8*(l>>4), col (l&15)

template <int K, int NOUT>
__global__ __launch_bounds__(256) void k_gemm_wmma(
    const float* __restrict__ X, const float* __restrict__ W,
    float* __restrict__ H, int n) {
    constexpr int NT = NOUT / 16;

    const int wave   = threadIdx.x >> 5;
    const int lane   = threadIdx.x & 31;
    const int tile_m = blockIdx.x * 8 + wave;
    const int row0   = tile_m * 16;
    if (row0 >= n) return;

    const int half = lane >> 4;   // 0: K+{0,1}   1: K+{2,3}
    const int l16  = lane & 15;

    int arow = row0 + l16;
    if (arow >= n) arow = n - 1;  // clamp loads; stores are guarded
    const float* __restrict__ aptr = X + (size_t)arow * K + half * 2;

    v8f zero = {};
    v8f acc[NT];
#pragma unroll
    for (int t = 0; t < NT; ++t) acc[t] = zero;

#pragma unroll 4
    for (int kk = 0; kk < K; kk += 4) {
        v2f a;
        a.x = aptr[kk];
        a.y = aptr[kk + 1];
        const float* __restrict__ w0 = W + (size_t)(kk + half * 2) * NOUT;
#pragma unroll
        for (int t = 0; t < NT; ++t) {
            v2f b;
            b.x = w0[t * 16 + l16];
            b.y = w0[NOUT + t * 16 + l16];
            acc[t] = __builtin_amdgcn_wmma_f32_16x16x4_f32(
                false, a, false, b, (short)0, acc[t], false, false);
        }
    }

#pragma unroll
    for (int t = 0; t < NT; ++t) {
#pragma unroll
        for (int v = 0; v < 8; ++v) {
            int r = row0 + v + half * 8;
            if (r < n) H[(size_t)r * NOUT + t * 16 + l16] = acc[t][v];
        }
    }
}

// ----------------------- bias broadcast / relu -----------------------------

template <int F>
__global__ void k_init_bias(float* __restrict__ out,
                            const float* __restrict__ b, int n) {
    long long i   = (long long)blockIdx.x * blockDim.x + threadIdx.x;
    long long tot = (long long)n * F;
    if (i < tot) out[i] = b[(int)(i & (F - 1))];  // F is a power of two
}

__global__ void k_relu(float* __restrict__ x, long long n) {
    long long i = (long long)blockIdx.x * blockDim.x + threadIdx.x;
    if (i < n) x[i] = fmaxf(x[i], 0.0f);
}

// ------------------------- edge scatter-add --------------------------------
// One wave per item (edge or self-loop); each of the 32 lanes carries
// F/32 = VEC consecutive floats of h[src], scaled by dinv[src]*dinv[dst],
// scatter-added into out[dst] via global_atomic_add_f32.

template <int F, int VEC>
__global__ __launch_bounds__(256) void k_scatter(
    const float* __restrict__ h, const long long* __restrict__ src,
    const long long* __restrict__ dst, const float* __restrict__ dinv,
    float* __restrict__ out, int E, int total) {
    static_assert(F / VEC == 32, "one wave covers one feature row");
    const int wid  = (blockIdx.x * blockDim.x + threadIdx.x) >> 5;
    const int lane = threadIdx.x & 31;
    const int nw   = (gridDim.x * blockDim.x) >> 5;

    for (int item = wid; item < total; item += nw) {
        int s, d;
        if (item < E) {
            s = (int)src[item];
            d = (int)dst[item];
        } else {
            s = item - E;  // self-loop
            d = s;
        }

        // prefetch next item's gather row (gfx1250 global_prefetch_b8)
        int nxt = item + nw;
        if (nxt < total) {
            int sn = (nxt < E) ? (int)src[nxt] : nxt - E;
            __builtin_prefetch(h + (size_t)sn * F + lane * VEC, 0, 1);
        }

        const float norm = dinv[s] * dinv[d];
        const float* __restrict__ hp = h + (size_t)s * F + lane * VEC;
        float* __restrict__ op       = out + (size_t)d * F + lane * VEC;

        float v[VEC];
        if constexpr (VEC == 4) {
            float4 t = *reinterpret_cast<const float4*>(hp);
            v[0] = t.x; v[1] = t.y; v[2] = t.z; v[3] = t.w;
        } else {
            float2 t = *reinterpret_cast<const float2*>(hp);
            v[0] = t.x; v[1] = t.y;
        }
#pragma unroll
        for (int j = 0; j < VEC; ++j) atomicAdd(&op[j], v[j] * norm);
    }
}

// ------------------------------- launcher ----------------------------------

extern "C" void kernel_launch(void* const* d_in, const int* in_sizes, int n_in,
                              void* d_out, int out_size, void* d_ws,
                              size_t ws_size, hipStream_t stream) {
    const float*     x   = (const float*)d_in[0];
    const long long* ei  = (const long long*)d_in[1];  // int64 (2, E)
    const float*     W1  = (const float*)d_in[2];
    const float*     b1  = (const float*)d_in[3];
    const float*     W2  = (const float*)d_in[4];
    const float*     b2  = (const float*)d_in[5];
    float*           out = (float*)d_out;

    const int N = in_sizes[0] / IN_CH;
    const int E = in_sizes[1] / 2;
    const long long* src = ei;
    const long long* dst = ei + E;
    const int total = E + N;  // edges + self-loops

    // workspace layout: dinv[N] | bufA[N*HID] | bufB[N*HID]
    float* dinv = (float*)d_ws;
    float* bufA = dinv + N;                  // H1 pre-agg, later H2 pre-agg
    float* bufB = bufA + (size_t)N * HID;    // agg1 -> relu'd h1

    const int thr = 256;
    // --- normalization ---
    k_init_deg<<<(N + thr - 1) / thr, thr, 0, stream>>>(dinv, N);
    k_count_deg<<<(E + thr - 1) / thr, thr, 0, stream>>>(dst, dinv, E);
    k_deg_to_dinv<<<(N + thr - 1) / thr, thr, 0, stream>>>(dinv, N);

    const int tiles_m = (N + 15) / 16;
    const int gemm_bx = (tiles_m + 7) / 8;

    // --- layer 1 ---
    k_gemm_wmma<IN_CH, HID><<<gemm_bx, thr, 0, stream>>>(x, W1, bufA, N);
    {
        long long tot = (long long)N * HID;
        k_init_bias<HID><<<(int)((tot + thr - 1) / thr), thr, 0, stream>>>(bufB, b1, N);
        k_scatter<HID, 4><<<4096, thr, 0, stream>>>(bufA, src, dst, dinv, bufB, E, total);
        k_relu<<<(int)((tot + thr - 1) / thr), thr, 0, stream>>>(bufB, tot);
    }

    // --- layer 2 ---
    k_gemm_wmma<HID, OUT_CH><<<gemm_bx, thr, 0, stream>>>(bufB, W2, bufA, N);
    {
        long long tot = (long long)N * OUT_CH;
        k_init_bias<OUT_CH><<<(int)((tot + thr - 1) / thr), thr, 0, stream>>>(out, b2, N);
        k_scatter<OUT_CH, 2><<<4096, thr, 0, stream>>>(bufA, src, dst, dinv, out, E, total);
    }
}